// CIFlow_base_54941221650933
// MI455X (gfx1250) — compile-verified
//
#include <hip/hip_runtime.h>
#include <stdint.h>

// ---------------------------------------------------------------------------
// CIFlow forward on MI455X (gfx1250, wave32).
// - Dominant GEMMs on V_WMMA_F32_16X16X4_F32 (full fp32: workload is HBM-bound
//   at ~75MB traffic / 1.6 GFLOP, so low-precision WMMA buys nothing and the
//   reference is strict f32).
// - H tile staging in kernel 1 uses the Tensor Data Mover (tensor_load_to_lds)
//   with the D# pad feature generating the bank-conflict-free 132-float LDS
//   row stride directly, waited on via s_wait_tensorcnt.
// ---------------------------------------------------------------------------

#define N_NODES   64000
#define N_GRAPHS  512
#define NODES_PG  125
#define HIDDEN    128
#define CLUSTERS  10
#define S_HID     69
#define S_HID_P   80     // padded to 5 x 16 WMMA tiles
#define P_HID     8
#define NUM_F     16

typedef float v2f __attribute__((ext_vector_type(2)));
typedef float v8f __attribute__((ext_vector_type(8)));

#if defined(__has_builtin)
#if __has_builtin(__builtin_amdgcn_tensor_load_to_lds)
#define HAVE_TDM 1
#endif
#endif
#ifndef HAVE_TDM
#define HAVE_TDM 0
#endif

#if HAVE_TDM
typedef unsigned int u32x4 __attribute__((ext_vector_type(4)));
typedef int          i32x4 __attribute__((ext_vector_type(4)));
typedef int          i32x8 __attribute__((ext_vector_type(8)));

// 2D TDM load: `rows` x `row_elems` f32 tile (row stride `row_stride_elems`)
// from global into LDS at byte offset lds_byte_addr, optionally inserting
// pad_amount DWORDs of LDS padding every (1<<pad_interval_code)*2 DWORDs.
// D# bitfields per CDNA5 ISA ch.8 (group0 128b, group1 256b; groups 2/3 zero
// for a 2D tensor).
__device__ inline void tdm_load_2d_f32(uint32_t lds_byte_addr, const void* gaddr,
                                       uint32_t rows, uint32_t row_elems,
                                       uint32_t row_stride_elems,
                                       uint32_t pad_interval_code,
                                       uint32_t pad_amount_dw_m1,
                                       bool pad_en)
{
    const uint64_t ga = (uint64_t)(uintptr_t)gaddr;
    u32x4 g0;
    g0[0] = 1u;                                            // count=1, user mode
    g0[1] = lds_byte_addr;                                 // lds_addr
    g0[2] = (uint32_t)(ga & 0xffffffffu);                  // global_addr[31:0]
    g0[3] = (uint32_t)((ga >> 32) & 0x01ffffffu)           // global_addr[56:32]
          | (2u << 30);                                    // type=2 ("image")

    uint32_t w0 = (2u << 16);                              // data_size = 4 bytes
    if (pad_en)
        w0 |= (1u << 20) | (pad_interval_code << 22) | (pad_amount_dw_m1 << 25);
    i32x8 g1;
    g1[0] = (int)w0;                                       // mask=0, size, pad
    g1[1] = (int)((row_elems & 0xffffu) << 16);            // tensor_dim0[15:0]
    g1[2] = (int)(((row_elems >> 16) & 0xffffu)            // tensor_dim0[31:16]
          | ((rows & 0xffffu) << 16));                     // tensor_dim1[15:0]
    g1[3] = (int)(((rows >> 16) & 0xffffu)                 // tensor_dim1[31:16]
          | ((row_elems & 0xffffu) << 16));                // tile_dim0
    g1[4] = (int)(rows & 0xffffu);                         // tile_dim1 (dim2=0)
    g1[5] = (int)row_stride_elems;                         // dim0_stride[31:0]
    g1[6] = 0;                                             // stride hi, dim1_stride lo
    g1[7] = 0;                                             // dim1_stride hi

    i32x4 z4 = {0, 0, 0, 0};
#if __clang_major__ >= 23
    i32x8 z8 = {0, 0, 0, 0, 0, 0, 0, 0};
    __builtin_amdgcn_tensor_load_to_lds(g0, g1, z4, z4, z8, 0);
#else
    __builtin_amdgcn_tensor_load_to_lds(g0, g1, z4, z4, 0);
#endif
}
#endif // HAVE_TDM

// ---------------------------------------------------------------------------
// Kernel 1: S_all = softmax(relu(H @ w1 + b1) @ w2 + b2)
// Block = 128 threads = 4 waves; each wave computes a 16-node tile.
// A-frag (16x4 f32): lane L -> M = L%16, K = k0 + 2*(L/16) + {0,1}
// B-frag (4x16 f32): lane L -> N = L%16, K = k0 + 2*(L/16) + {0,1}
// D (16x16 f32):     lane L, vgpr r -> M = r + 8*(L/16), N = L%16
// ---------------------------------------------------------------------------
__global__ __launch_bounds__(128) void s_assign_kernel(
    const float* __restrict__ H,  const float* __restrict__ w1,
    const float* __restrict__ b1, const float* __restrict__ w2,
    const float* __restrict__ b2, float* __restrict__ S)
{
    __shared__ float sH[64][132];   // 64 nodes x 128 (stride 132: bank-conflict-free A reads)
    __shared__ float sT[64][84];    // relu(H@w1+b1), zero-padded cols 69..79
    __shared__ float sS[64][17];    // stage-2 logits

    const int tid       = threadIdx.x;
    const int node_base = blockIdx.x * 64;

#if HAVE_TDM
    // One TDM DMA per block (wave 0 issues; TDM ignores EXEC). The pad fields
    // (every 128 DWORDs insert 4 DWORDs => LDS row stride 132 floats) write
    // the bank-conflict-free layout directly.
    if (tid < 32) {
        tdm_load_2d_f32((uint32_t)(uintptr_t)&sH[0][0],
                        H + (size_t)node_base * HIDDEN,
                        /*rows=*/64, /*row_elems=*/HIDDEN,
                        /*row_stride=*/HIDDEN,
                        /*pad_interval: 1<<6 * 8B = 128 DWORDs*/ 6u,
                        /*pad_amount: code 3 = 4 DWORDs*/ 3u,
                        /*pad_en=*/true);
        __builtin_amdgcn_s_wait_tensorcnt(0);
    }
#else
    // cooperative coalesced load of the 64x128 H tile (float4)
    for (int i = tid; i < 64 * 32; i += 128) {
        const int r  = i >> 5;
        const int c4 = i & 31;
        const float4 v =
            reinterpret_cast<const float4*>(H + (size_t)(node_base + r) * HIDDEN)[c4];
        sH[r][c4 * 4 + 0] = v.x;
        sH[r][c4 * 4 + 1] = v.y;
        sH[r][c4 * 4 + 2] = v.z;
        sH[r][c4 * 4 + 3] = v.w;
    }
#endif
    __syncthreads();

    const int wave = tid >> 5;
    const int lane = tid & 31;
    const int ln   = lane & 15;
    const int half = lane >> 4;
    const int row0 = wave * 16;

    // ---- stage 1: T = relu(H_tile @ w1 + b1), N padded 69 -> 80 ------------
    for (int j = 0; j < 5; ++j) {
        const int n0 = j * 16;
        const int n  = n0 + ln;
        const float bias = (n < S_HID) ? b1[n] : 0.0f;
        v8f acc;
        #pragma unroll
        for (int r = 0; r < 8; ++r) acc[r] = bias;

        for (int kk = 0; kk < 32; ++kk) {
            const int k = kk * 4 + half * 2;
            v2f a = { sH[row0 + ln][k], sH[row0 + ln][k + 1] };
            v2f b;
            b.x = (n < S_HID) ? w1[(size_t)k       * S_HID + n] : 0.0f;
            b.y = (n < S_HID) ? w1[(size_t)(k + 1) * S_HID + n] : 0.0f;
            acc = __builtin_amdgcn_wmma_f32_16x16x4_f32(
                false, a, false, b, (short)0, acc, false, false);
        }
        #pragma unroll
        for (int r = 0; r < 8; ++r) {
            float v = acc[r];
            sT[row0 + r + half * 8][n0 + ln] = v > 0.0f ? v : 0.0f;
        }
    }
    __syncthreads();

    // ---- stage 2: logits = T @ w2 + b2 ; K padded 69 -> 80, N padded -> 16 -
    {
        const float bias2 = (ln < CLUSTERS) ? b2[ln] : 0.0f;
        v8f acc;
        #pragma unroll
        for (int r = 0; r < 8; ++r) acc[r] = bias2;

        for (int kk = 0; kk < 20; ++kk) {
            const int k = kk * 4 + half * 2;
            v2f a = { sT[row0 + ln][k], sT[row0 + ln][k + 1] };
            v2f b;
            b.x = (k     < S_HID && ln < CLUSTERS) ? w2[(size_t)k       * CLUSTERS + ln] : 0.0f;
            b.y = (k + 1 < S_HID && ln < CLUSTERS) ? w2[(size_t)(k + 1) * CLUSTERS + ln] : 0.0f;
            acc = __builtin_amdgcn_wmma_f32_16x16x4_f32(
                false, a, false, b, (short)0, acc, false, false);
        }
        #pragma unroll
        for (int r = 0; r < 8; ++r)
            sS[row0 + r + half * 8][ln] = acc[r];
    }
    __syncthreads();

    // ---- row softmax over 10 clusters: thread t handles row t --------------
    if (tid < 64) {
        float m = sS[tid][0];
        #pragma unroll
        for (int c = 1; c < CLUSTERS; ++c) m = fmaxf(m, sS[tid][c]);
        float e[CLUSTERS], sum = 0.0f;
        #pragma unroll
        for (int c = 0; c < CLUSTERS; ++c) { e[c] = expf(sS[tid][c] - m); sum += e[c]; }
        const float inv = 1.0f / sum;
        float* out = S + (size_t)(node_base + tid) * CLUSTERS;
        #pragma unroll
        for (int c = 0; c < CLUSTERS; ++c) out[c] = e[c] * inv;
    }
}

// ---------------------------------------------------------------------------
// Kernel 2: E[g] = S_g^T @ H_g   ([10x125] @ [125x128], M pad 16, K pad 128)
// One block per graph; 8 waves = 8 feature tiles of 16.
// ---------------------------------------------------------------------------
__global__ __launch_bounds__(256) void e_pool_kernel(
    const float* __restrict__ S, const float* __restrict__ H,
    float* __restrict__ E)
{
    __shared__ float sA[16][132];   // S^T, zero padded: [cluster][node]

    const int g   = blockIdx.x;
    const int tid = threadIdx.x;

    float* sAf = &sA[0][0];
    for (int i = tid; i < 16 * 132; i += 256) sAf[i] = 0.0f;
    __syncthreads();

    const size_t sbase = (size_t)g * NODES_PG * CLUSTERS;
    for (int i = tid; i < NODES_PG * CLUSTERS; i += 256) {
        const int node = i / CLUSTERS;
        const int c    = i - node * CLUSTERS;
        sA[c][node] = S[sbase + i];
    }
    __syncthreads();

    const int wave = tid >> 5;
    const int lane = tid & 31;
    const int ln   = lane & 15;
    const int half = lane >> 4;
    const int n0   = wave * 16;
    const size_t hbase = (size_t)g * NODES_PG * HIDDEN;

    __builtin_prefetch(H + hbase + n0 + ln, 0, 3);   // global_prefetch into GL2

    v8f acc;
    #pragma unroll
    for (int r = 0; r < 8; ++r) acc[r] = 0.0f;

    for (int kk = 0; kk < 32; ++kk) {
        const int k = kk * 4 + half * 2;            // node index (K)
        v2f a = { sA[ln][k], sA[ln][k + 1] };
        v2f b;
        b.x = (k     < NODES_PG) ? H[hbase + (size_t)k       * HIDDEN + n0 + ln] : 0.0f;
        b.y = (k + 1 < NODES_PG) ? H[hbase + (size_t)(k + 1) * HIDDEN + n0 + ln] : 0.0f;
        acc = __builtin_amdgcn_wmma_f32_16x16x4_f32(
            false, a, false, b, (short)0, acc, false, false);
    }
    #pragma unroll
    for (int r = 0; r < 8; ++r) {
        const int c = r + half * 8;
        if (c < CLUSTERS)
            E[(size_t)g * (CLUSTERS * HIDDEN) + (size_t)c * HIDDEN + n0 + ln] = acc[r];
    }
}

// ---------------------------------------------------------------------------
// JAX threefry2x32-20 (key = jax.random.key(42) -> data [0, 42]) to reproduce
// jax.random.categorical's Gumbel trick for shape (512, 2).
// ---------------------------------------------------------------------------
__device__ inline uint32_t rotl32(uint32_t v, int n) { return (v << n) | (v >> (32 - n)); }

__device__ inline void threefry2x32_42(uint32_t x0, uint32_t x1,
                                       uint32_t* o0, uint32_t* o1)
{
    const uint32_t ks[3] = {0u, 42u, 0u ^ 42u ^ 0x1BD11BDAu};
    const int R0[4] = {13, 15, 26, 6};
    const int R1[4] = {17, 29, 16, 24};
    x0 += ks[0]; x1 += ks[1];
    #pragma unroll
    for (int i = 0; i < 5; ++i) {
        const int* rc = (i & 1) ? R1 : R0;
        #pragma unroll
        for (int j = 0; j < 4; ++j) { x0 += x1; x1 = rotl32(x1, rc[j]); x1 ^= x0; }
        x0 += ks[(i + 1) % 3];
        x1 += ks[(i + 2) % 3] + (uint32_t)(i + 1);
    }
    *o0 = x0; *o1 = x1;
}

// JAX threefry_random_bits for 1024 words: counts split [0:512],[512:1024].
__device__ inline uint32_t jax_bits(uint32_t flat)
{
    uint32_t o0, o1;
    if (flat < 512u) { threefry2x32_42(flat, flat + 512u, &o0, &o1); return o0; }
    threefry2x32_42(flat - 512u, flat, &o0, &o1);
    return o1;
}

__device__ inline float gumbel_from_bits(uint32_t bits)
{
    const float tiny = 1.17549435e-38f;
    float u = __uint_as_float(0x3f800000u | (bits >> 9)) - 1.0f;  // [0,1)
    u = u * (1.0f - tiny) + tiny;                                  // jax minval=tiny
    return -logf(-logf(u));
}

// ---------------------------------------------------------------------------
// Kernel 3: pred1 = E.reshape(B,1280) @ wf + bf ; softmax ; categorical sample
// One block (128 threads) per graph.
// ---------------------------------------------------------------------------
__global__ __launch_bounds__(128) void pred1_sample_kernel(
    const float* __restrict__ E,  const float* __restrict__ wf,
    const float* __restrict__ bf, const int* __restrict__ targets,
    float* __restrict__ pred1, float* __restrict__ ind,
    int* __restrict__ sample_ws)
{
    __shared__ float red[2][128];
    const int g   = blockIdx.x;
    const int tid = threadIdx.x;

    float a0 = 0.0f, a1 = 0.0f;
    const float* e = E + (size_t)g * (CLUSTERS * HIDDEN);
    for (int i = tid; i < CLUSTERS * HIDDEN; i += 128) {
        const float v = e[i];
        a0 += v * wf[2 * i + 0];
        a1 += v * wf[2 * i + 1];
    }
    red[0][tid] = a0; red[1][tid] = a1;
    __syncthreads();
    for (int s = 64; s > 0; s >>= 1) {
        if (tid < s) { red[0][tid] += red[0][tid + s]; red[1][tid] += red[1][tid + s]; }
        __syncthreads();
    }

    if (tid == 0) {
        const float l0 = red[0][0] + bf[0];
        const float l1 = red[1][0] + bf[1];
        pred1[2 * g + 0] = l0;
        pred1[2 * g + 1] = l1;
        const float m  = fmaxf(l0, l1);
        const float e0 = expf(l0 - m), e1 = expf(l1 - m);
        const float p0 = e0 / (e0 + e1), p1 = e1 / (e0 + e1);
        const float z0 = logf(p0 + 1e-12f) + gumbel_from_bits(jax_bits(2u * g + 0u));
        const float z1 = logf(p1 + 1e-12f) + gumbel_from_bits(jax_bits(2u * g + 1u));
        const int smp  = (z1 > z0) ? 1 : 0;
        sample_ws[g] = smp;
        ind[g] = (smp == targets[g]) ? 1.0f : 0.0f;
    }
}

// ---------------------------------------------------------------------------
// Kernel 4: Q = softmax(relu(E @ p1 + pb1) @ p2 + pb2)  over NUM_F
// One thread per (graph, cluster) row; 5120 rows.
// ---------------------------------------------------------------------------
__global__ __launch_bounds__(128) void q_kernel(
    const float* __restrict__ E,  const float* __restrict__ p1,
    const float* __restrict__ pb1, const float* __restrict__ p2,
    const float* __restrict__ pb2, float* __restrict__ Q)
{
    const int row = blockIdx.x * blockDim.x + threadIdx.x;
    if (row >= N_GRAPHS * CLUSTERS) return;

    const float* e = E + (size_t)row * HIDDEN;
    float h[P_HID];
    #pragma unroll
    for (int j = 0; j < P_HID; ++j) h[j] = pb1[j];
    for (int d = 0; d < HIDDEN; ++d) {
        const float v = e[d];
        #pragma unroll
        for (int j = 0; j < P_HID; ++j) h[j] += v * p1[d * P_HID + j];
    }
    #pragma unroll
    for (int j = 0; j < P_HID; ++j) h[j] = fmaxf(h[j], 0.0f);

    float q[NUM_F];
    #pragma unroll
    for (int f = 0; f < NUM_F; ++f) q[f] = pb2[f];
    #pragma unroll
    for (int j = 0; j < P_HID; ++j)
        #pragma unroll
        for (int f = 0; f < NUM_F; ++f) q[f] += h[j] * p2[j * NUM_F + f];

    float m = q[0];
    #pragma unroll
    for (int f = 1; f < NUM_F; ++f) m = fmaxf(m, q[f]);
    float sum = 0.0f;
    #pragma unroll
    for (int f = 0; f < NUM_F; ++f) { q[f] = expf(q[f] - m); sum += q[f]; }
    const float inv = 1.0f / sum;
    float* out = Q + (size_t)row * NUM_F;
    #pragma unroll
    for (int f = 0; f < NUM_F; ++f) out[f] = q[f] * inv;
}

// ---------------------------------------------------------------------------
// Kernel 5: correlation mask -> feature_mask -> pred2. One block per graph.
// sample_cat_oh is exactly one_hot(sample), so corr_mask = correlation[sample].
// ---------------------------------------------------------------------------
__global__ __launch_bounds__(128) void pred2_kernel(
    const float* __restrict__ E,  const float* __restrict__ Q,
    const float* __restrict__ wf, const float* __restrict__ bf,
    const float* __restrict__ ci, const float* __restrict__ cj,
    const int* __restrict__ sample_ws, float* __restrict__ pred2)
{
    __shared__ float cm[NUM_F];
    __shared__ float fm[CLUSTERS];
    __shared__ float red[2][128];

    const int g   = blockIdx.x;
    const int tid = threadIdx.x;
    const int smp = sample_ws[g];

    if (tid < NUM_F) {
        // inter: softmax over classes (axis 0) at filter tid
        const float a0 = ci[tid], a1 = ci[NUM_F + tid];
        const float m01 = fmaxf(a0, a1);
        const float ee0 = expf(a0 - m01), ee1 = expf(a1 - m01);
        const float inter = ((smp == 0) ? ee0 : ee1) / (ee0 + ee1);
        // intra: softmax over filters (axis 1) in row smp
        float rm = cj[smp * NUM_F];
        for (int f = 1; f < NUM_F; ++f) rm = fmaxf(rm, cj[smp * NUM_F + f]);
        float rs = 0.0f;
        for (int f = 0; f < NUM_F; ++f) rs += expf(cj[smp * NUM_F + f] - rm);
        const float intra = expf(cj[smp * NUM_F + tid] - rm) / rs;
        cm[tid] = inter * intra;
    }
    __syncthreads();

    if (tid < CLUSTERS) {
        float s = 0.0f;
        const float* qrow = Q + (size_t)g * CLUSTERS * NUM_F + (size_t)tid * NUM_F;
        #pragma unroll
        for (int f = 0; f < NUM_F; ++f) s += qrow[f] * cm[f];
        fm[tid] = s;
    }
    __syncthreads();

    float a0 = 0.0f, a1 = 0.0f;
    const float* e = E + (size_t)g * (CLUSTERS * HIDDEN);
    for (int i = tid; i < CLUSTERS * HIDDEN; i += 128) {
        const int c   = i >> 7;
        const float v = fm[c] * e[i];
        a0 += v * wf[2 * i + 0];
        a1 += v * wf[2 * i + 1];
    }
    red[0][tid] = a0; red[1][tid] = a1;
    __syncthreads();
    for (int s = 64; s > 0; s >>= 1) {
        if (tid < s) { red[0][tid] += red[0][tid + s]; red[1][tid] += red[1][tid + s]; }
        __syncthreads();
    }
    if (tid == 0) {
        pred2[2 * g + 0] = red[0][0] + bf[0];
        pred2[2 * g + 1] = red[1][0] + bf[1];
    }
}

// ---------------------------------------------------------------------------
extern "C" void kernel_launch(void* const* d_in, const int* in_sizes, int n_in,
                              void* d_out, int out_size, void* d_ws, size_t ws_size,
                              hipStream_t stream)
{
    (void)in_sizes; (void)n_in; (void)out_size; (void)ws_size;

    const float* H       = (const float*)d_in[0];
    /* d_in[1] = batch: implied by fixed 125 nodes/graph */
    const int*   targets = (const int*)  d_in[2];
    const float* w1      = (const float*)d_in[3];
    const float* b1      = (const float*)d_in[4];
    const float* w2      = (const float*)d_in[5];
    const float* b2      = (const float*)d_in[6];
    const float* wf      = (const float*)d_in[7];
    const float* bf      = (const float*)d_in[8];
    const float* p1      = (const float*)d_in[9];
    const float* pb1     = (const float*)d_in[10];
    const float* p2      = (const float*)d_in[11];
    const float* pb2     = (const float*)d_in[12];
    const float* ci      = (const float*)d_in[13];
    const float* cj      = (const float*)d_in[14];

    float* out   = (float*)d_out;
    float* pred1 = out;                                     //  512*2
    float* pred2 = out + 1024;                              //  512*2
    float* ind   = out + 2048;                              //  512
    float* S     = out + 2560;                              //  64000*10
    float* E     = out + 2560 + 640000;                     //  512*10*128
    float* Q     = out + 2560 + 640000 + 655360;            //  512*10*16

    int* sample_ws = (int*)d_ws;

    s_assign_kernel    <<<N_NODES / 64, 128, 0, stream>>>(H, w1, b1, w2, b2, S);
    e_pool_kernel      <<<N_GRAPHS,     256, 0, stream>>>(S, H, E);
    pred1_sample_kernel<<<N_GRAPHS,     128, 0, stream>>>(E, wf, bf, targets, pred1, ind, sample_ws);
    q_kernel           <<<(N_GRAPHS * CLUSTERS + 127) / 128, 128, 0, stream>>>(E, p1, pb1, p2, pb2, Q);
    pred2_kernel       <<<N_GRAPHS,     128, 0, stream>>>(E, Q, wf, bf, ci, cj, sample_ws, pred2);
}